// Gaussian2DMinimal_89395449299585
// MI455X (gfx1250) — compile-verified
//
#include <hip/hip_runtime.h>
#include <hip/hip_bf16.h>

// Problem geometry (fixed by the reference).
#define NG   512
#define HIMG 256
#define WIMG 256
#define HW   (HIMG * WIMG)
#define PP   12   // floats per gaussian param record: mx,my,a,b,c,opac,r,g,bc,pad,pad,pad

typedef __attribute__((ext_vector_type(2))) float v2f;
typedef __attribute__((ext_vector_type(8))) float v8f;
typedef int v4i_ __attribute__((__vector_size__(16)));

#if __has_builtin(__builtin_amdgcn_wmma_f32_16x16x4_f32)
#define HAS_WMMA4 1
#else
#define HAS_WMMA4 0
#endif

#if __has_builtin(__builtin_amdgcn_global_load_async_to_lds_b128) && \
    __has_builtin(__builtin_amdgcn_s_wait_asynccnt)
#define HAS_ASYNC 1
#else
#define HAS_ASYNC 0
#endif

__device__ __forceinline__ float fast_sigmoid(float x) {
  return 1.0f / (1.0f + __expf(-x));
}
__device__ __forceinline__ float softplusf(float x) {
  // numerically stable softplus
  return fmaxf(x, 0.0f) + log1pf(__expf(-fabsf(x)));
}

// -------------------- Kernel 1: per-gaussian preprocessing --------------------
__global__ void gs_pre(const float* __restrict__ means, const float* __restrict__ chol,
                       const float* __restrict__ rgbl, const float* __restrict__ opl,
                       float* __restrict__ P) {
  int i = blockIdx.x * blockDim.x + threadIdx.x;
  if (i >= NG) return;
  float L11 = softplusf(chol[3 * i + 0]);
  float L21 = chol[3 * i + 1];
  float L22 = softplusf(chol[3 * i + 2]);
  float s00 = L11 * L11;
  float s01 = L11 * L21;
  float s11 = fmaf(L21, L21, L22 * L22);
  float d   = L11 * L22;
  float inv = 1.0f / (d * d);
  float* p = P + i * PP;
  p[0] = means[2 * i + 0];
  p[1] = means[2 * i + 1];
  p[2] = s11 * inv;        // a
  p[3] = -s01 * inv;       // b
  p[4] = s00 * inv;        // c
  p[5] = fast_sigmoid(opl[i]);
  p[6] = fast_sigmoid(rgbl[3 * i + 0]);
  p[7] = fast_sigmoid(rgbl[3 * i + 1]);
  p[8] = fast_sigmoid(rgbl[3 * i + 2]);
  p[9] = 0.f; p[10] = 0.f; p[11] = 0.f;
}

// Cross-half (lane ^ 16) value exchange inside a wave32.
__device__ __forceinline__ float xswap16(float v) {
#if __has_builtin(__builtin_amdgcn_permlanex16)
  int u = __float_as_int(v);
  u = __builtin_amdgcn_permlanex16(u, u, 0x76543210, (int)0xFEDCBA98, false, false);
  return __int_as_float(u);
#else
  int src = (((int)(threadIdx.x & 31)) ^ 16) << 2;
  return __int_as_float(__builtin_amdgcn_ds_bpermute(src, __float_as_int(v)));
#endif
}

// -------------------- Kernel 2: render --------------------
// 256 threads = 8 waves; each wave owns 32 consecutive pixels.
// Per gaussian: compute per-pixel weights w0..w3 (vis, vis*gx, vis*gy,
// vis*(gx*gy-b)), then accumulate the 16(pixel) x 16(channel) output block
// with V_WMMA_F32_16X16X4_F32 (K = 4 kinds per gaussian), 2 WMMAs per wave.
__global__ __launch_bounds__(256) void gs_render(const float* __restrict__ P,
                                                 float* __restrict__ out) {
  __shared__ __align__(16) float sp[NG * PP];  // 24 KB param table
  const int tid = threadIdx.x;

  // ---- stage params to LDS (gfx1250 async-to-LDS path when available) ----
  {
    const float4* g4 = (const float4*)P;
    float4* l4 = (float4*)sp;
#if HAS_ASYNC
#pragma unroll
    for (int i = 0; i < (NG * PP / 4) / 256; ++i) {
      int idx = i * 256 + tid;
      __builtin_amdgcn_global_load_async_to_lds_b128(
          (__attribute__((address_space(1))) v4i_*)(g4 + idx),
          (__attribute__((address_space(3))) v4i_*)(l4 + idx), 0, 0);
    }
    __builtin_amdgcn_s_wait_asynccnt(0);
#else
    for (int idx = tid; idx < NG * PP / 4; idx += 256) l4[idx] = g4[idx];
#endif
  }
  __syncthreads();

  const int lane    = tid & 31;
  const int wave    = tid >> 5;
  const int pixBase = blockIdx.x * 256 + wave * 32;
  const int myPix   = pixBase + lane;
  const float px = (float)(myPix % WIMG) + 0.5f;
  const float py = (float)(myPix / WIMG) + 0.5f;

  // Output sections (flat concatenation in return order).
  float* renderp = out;                 // (3,H,W)
  float* imgp    = out + 3 * HW;        // (H,W,3)
  float* wsump   = out + 6 * HW;        // (H,W)
  float* dxp     = out + 7 * HW;        // (H,W,3)
  float* dyp     = out + 10 * HW;       // (H,W,3)
  float* dxyp    = out + 13 * HW;       // (H,W,3)

  float T = 1.0f;

#if HAS_WMMA4
  const int  chan = lane & 15;
  const bool hiH  = lane >= 16;
  // Per-lane constant masks to build the per-gaussian B matrix (4 kinds x 16 ch):
  //  row0(vis):   r@0 g@1 b@2 1@3 | row1(gx): r@4 g@5 b@6
  //  row2(gy):    r@7 g@8 b@9     | row3(gxgy-b): r@10 g@11 b@12
  // VGPR b0 holds rows {0,2} (lo/hi half), b1 holds rows {1,3}.
  const float c0r = ((!hiH && chan == 0) || (hiH && chan == 7))  ? 1.f : 0.f;
  const float c0g = ((!hiH && chan == 1) || (hiH && chan == 8))  ? 1.f : 0.f;
  const float c0b = ((!hiH && chan == 2) || (hiH && chan == 9))  ? 1.f : 0.f;
  const float c0one = (!hiH && chan == 3) ? 1.f : 0.f;
  const float c1r = ((!hiH && chan == 4) || (hiH && chan == 10)) ? 1.f : 0.f;
  const float c1g = ((!hiH && chan == 5) || (hiH && chan == 11)) ? 1.f : 0.f;
  const float c1b = ((!hiH && chan == 6) || (hiH && chan == 12)) ? 1.f : 0.f;

  v8f dlo = {};  // pixels pixBase+0..15  (16 x 16ch, f32 accum)
  v8f dhi = {};  // pixels pixBase+16..31

  for (int n = 0; n < NG; ++n) {
    const float4 q0 = *(const float4*)(sp + n * PP);      // mx,my,a,b
    const float4 q1 = *(const float4*)(sp + n * PP + 4);  // c,opac,r,g
    const float  bc = sp[n * PP + 8];                     // blue

    float dX = q0.x - px, dY = q0.y - py;
    float gx = fmaf(q0.z, dX, q0.w * dY);   // a*dX + b*dY
    float gy = fmaf(q0.w, dX, q1.x * dY);   // b*dX + c*dY
    float sig = 0.5f * fmaf(dY, gy, dX * gx);
    float alpha = fminf(q1.y * __expf(-sig), 0.999f);
    float vis = alpha * T;
    T = fmaf(-alpha, T, T);                 // T *= (1 - alpha), old T used above
    float w0 = vis;
    float w1 = vis * gx;
    float w2 = vis * gy;
    float w3 = vis * fmaf(gx, gy, -q0.w);

    // B operand: per-gaussian channel matrix (same for both WMMAs).
    float b0 = fmaf(q1.z, c0r, fmaf(q1.w, c0g, fmaf(bc, c0b, c0one)));
    float b1 = fmaf(q1.z, c1r, fmaf(q1.w, c1g, bc * c1b));
    v2f B = {b0, b1};

    // A operands: A[m=pixel, k=kind]; VGPR0 = K0(lo)/K2(hi), VGPR1 = K1/K3.
    float x0 = xswap16(w0);
    float x1 = xswap16(w1);
    float x2 = xswap16(w2);
    float x3 = xswap16(w3);
    v2f Alo = {hiH ? x2 : w0, hiH ? x3 : w1};   // pixels 0..15 of this wave
    v2f Ahi = {hiH ? w2 : x0, hiH ? w3 : x1};   // pixels 16..31

    dlo = __builtin_amdgcn_wmma_f32_16x16x4_f32(false, Alo, false, B, (short)0,
                                                dlo, false, false);
    dhi = __builtin_amdgcn_wmma_f32_16x16x4_f32(false, Ahi, false, B, (short)0,
                                                dhi, false, false);
  }

  // ---- store epilogue, branch-minimal ----
  // D layout: VGPR v, lanes 0-15 -> M=v, lanes 16-31 -> M=v+8; col = chan.
  // Resolve this lane's output section ONCE into (base, stride), then two
  // hoisted divergent regions do plain strided stores.
  const int rowOff = hiH ? 8 : 0;
  float* base;
  int stride;
  if (chan < 3)       { base = imgp + chan;        stride = 3; }
  else if (chan == 3) { base = wsump;              stride = 1; }
  else if (chan < 7)  { base = dxp + (chan - 4);   stride = 3; }
  else if (chan < 10) { base = dyp + (chan - 7);   stride = 3; }
  else                { base = dxyp + (chan - 10); stride = 3; }

  if (chan < 13) {
#pragma unroll
    for (int v = 0; v < 8; ++v) {
      int plo = pixBase + v + rowOff;
      base[plo * stride] = dlo[v];
      base[(plo + 16) * stride] = dhi[v];
    }
  }
  if (chan < 3) {  // extra transposed copy: render = (3,H,W)
    float* rp = renderp + chan * HW;
#pragma unroll
    for (int v = 0; v < 8; ++v) {
      int plo = pixBase + v + rowOff;
      rp[plo] = dlo[v];
      rp[plo + 16] = dhi[v];
    }
  }
#else
  // ---- scalar fallback: 13 per-pixel accumulators ----
  float acc[13];
#pragma unroll
  for (int k = 0; k < 13; ++k) acc[k] = 0.f;
  for (int n = 0; n < NG; ++n) {
    const float4 q0 = *(const float4*)(sp + n * PP);
    const float4 q1 = *(const float4*)(sp + n * PP + 4);
    const float  bc = sp[n * PP + 8];
    float dX = q0.x - px, dY = q0.y - py;
    float gx = fmaf(q0.z, dX, q0.w * dY);
    float gy = fmaf(q0.w, dX, q1.x * dY);
    float sig = 0.5f * fmaf(dY, gy, dX * gx);
    float alpha = fminf(q1.y * __expf(-sig), 0.999f);
    float vis = alpha * T;
    T = fmaf(-alpha, T, T);
    float w1 = vis * gx, w2 = vis * gy, w3 = vis * fmaf(gx, gy, -q0.w);
    acc[0] = fmaf(vis, q1.z, acc[0]); acc[1] = fmaf(vis, q1.w, acc[1]); acc[2] = fmaf(vis, bc, acc[2]);
    acc[3] += vis;
    acc[4] = fmaf(w1, q1.z, acc[4]); acc[5] = fmaf(w1, q1.w, acc[5]); acc[6] = fmaf(w1, bc, acc[6]);
    acc[7] = fmaf(w2, q1.z, acc[7]); acc[8] = fmaf(w2, q1.w, acc[8]); acc[9] = fmaf(w2, bc, acc[9]);
    acc[10] = fmaf(w3, q1.z, acc[10]); acc[11] = fmaf(w3, q1.w, acc[11]); acc[12] = fmaf(w3, bc, acc[12]);
  }
#pragma unroll
  for (int c = 0; c < 3; ++c) {
    imgp[myPix * 3 + c] = acc[c];
    renderp[c * HW + myPix] = acc[c];
    dxp[myPix * 3 + c]  = acc[4 + c];
    dyp[myPix * 3 + c]  = acc[7 + c];
    dxyp[myPix * 3 + c] = acc[10 + c];
  }
  wsump[myPix] = acc[3];
#endif
}

extern "C" void kernel_launch(void* const* d_in, const int* in_sizes, int n_in,
                              void* d_out, int out_size, void* d_ws, size_t ws_size,
                              hipStream_t stream) {
  const float* means = (const float*)d_in[0];
  const float* chol  = (const float*)d_in[1];
  const float* rgbl  = (const float*)d_in[2];
  const float* opl   = (const float*)d_in[3];
  float* params = (float*)d_ws;  // NG*PP*4 = 24 KB scratch

  gs_pre<<<(NG + 255) / 256, 256, 0, stream>>>(means, chol, rgbl, opl, params);
  gs_render<<<HW / 256, 256, 0, stream>>>(params, (float*)d_out);
}